// InterestCandidateAttention_7301444403688
// MI455X (gfx1250) — compile-verified
//
#include <hip/hip_runtime.h>
#include <hip/hip_bf16.h>

typedef float v2f __attribute__((ext_vector_type(2)));
typedef float v8f __attribute__((ext_vector_type(8)));
typedef int   v4i __attribute__((vector_size(16)));

typedef __attribute__((address_space(1))) v4i gas_v4i;
typedef __attribute__((address_space(3))) v4i las_v4i;

#define D_DIM 256
#define K_DIM 50
#define K_PAD 52          // pad to multiple of 4 for 16x16x4 WMMA K-chunks
#define ROW   272         // LDS row stride (floats); 272 % 64 = 16 -> lane-halves hit disjoint banks

#if defined(__gfx1250__) && __has_builtin(__builtin_amdgcn_global_load_async_to_lds_b128) && \
    __has_builtin(__builtin_amdgcn_s_wait_asynccnt)
#define USE_ASYNC_LDS 1
#else
#define USE_ASYNC_LDS 0
#endif

__global__ __launch_bounds__(256)
void ica_kernel(const float* __restrict__ interest,
                const float* __restrict__ cand,
                const long long* __restrict__ counts,
                float* __restrict__ out) {
    __shared__ float sV[K_PAD * ROW];   // interest tile, rows 50..51 zeroed
    __shared__ float sC[D_DIM];         // candidate row
    __shared__ float sA[K_DIM];         // attn scores
    __shared__ float sW[K_PAD];         // masked weights (padded with zeros)

    const int tid  = threadIdx.x;
    const int b    = blockIdx.x;
    const int lane = tid & 31;
    const int wave = tid >> 5;

    // ---- Phase 0: stage V (50x256 f32) + candidate into LDS, zero pad rows ----
    {
        const float4* g4 = (const float4*)(interest + (size_t)b * K_DIM * D_DIM);

        // Zero the two pad rows (512 floats = 128 float4) with plain LDS stores.
        if (tid < 128) {
            int k  = 50 + (tid >> 6);
            int d4 = tid & 63;
            float* p = &sV[k * ROW + d4 * 4];
            p[0] = 0.f; p[1] = 0.f; p[2] = 0.f; p[3] = 0.f;
        }

#if USE_ASYNC_LDS
        // HBM -> LDS direct, tracked by ASYNCcnt; no VGPR round trip.
        #pragma unroll
        for (int j = 0; j < 13; ++j) {              // 3200 float4 slots (50 rows)
            int idx = tid + 256 * j;
            if (idx < K_DIM * 64) {
                int k  = idx >> 6;
                int d4 = idx & 63;
                __builtin_amdgcn_global_load_async_to_lds_b128(
                    (gas_v4i*)(g4 + idx),
                    (las_v4i*)&sV[k * ROW + d4 * 4],
                    0, 0);
            }
        }
        sC[tid] = cand[(size_t)b * D_DIM + tid];
        __builtin_amdgcn_s_wait_asynccnt(0);
#else
        #pragma unroll
        for (int j = 0; j < 13; ++j) {
            int idx = tid + 256 * j;
            if (idx < K_DIM * 64) {
                int k  = idx >> 6;
                int d4 = idx & 63;
                float4 v = g4[idx];
                float* p = &sV[k * ROW + d4 * 4];
                p[0] = v.x; p[1] = v.y; p[2] = v.z; p[3] = v.w;
            }
        }
        sC[tid] = cand[(size_t)b * D_DIM + tid];
#endif
    }
    __syncthreads();

    // ---- Phase 1: attn[k] = dot(V[k,:], c) ; wave-parallel + shuffle reduce ----
    for (int k = wave; k < K_DIM; k += 8) {
        float s = 0.f;
        #pragma unroll
        for (int j = 0; j < 8; ++j)
            s = fmaf(sV[k * ROW + lane + 32 * j], sC[lane + 32 * j], s);
        #pragma unroll
        for (int off = 16; off > 0; off >>= 1)
            s += __shfl_xor(s, off, 32);
        if (lane == 0) sA[k] = s;
    }
    __syncthreads();

    // ---- Phase 2: rank-based top-dynK mask (stable descending order) ----
    if (tid < K_PAD) {
        float w = 0.f;
        if (tid < K_DIM) {
            long long c = counts[b];
            if (c < 1) c = 1;
            int x = (int)(5 * c);                              // 5..95, never a power of 2
            int dynK = 32 - __builtin_clz((unsigned)(x - 1));  // == ceil(log2(x))
            if (dynK < 1) dynK = 1;
            if (dynK > K_DIM) dynK = K_DIM;
            float ak = sA[tid];
            int rank = 0;
            #pragma unroll 1
            for (int j = 0; j < K_DIM; ++j) {
                float aj = sA[j];
                rank += (aj > ak) || (aj == ak && j < tid);
            }
            if (rank < dynK) w = ak;
        }
        sW[tid] = w;
    }
    __syncthreads();

    // ---- Phase 3: user[d] = sum_k w[k]*V[k,d] via v_wmma_f32_16x16x4_f32 ----
    // A[m][kk] = w[kk] (broadcast over M rows), B[kk][n] = V[kk][d0+n].
    // f32 A/B layout: lane-half selects K-half, VGPR selects K within half.
    {
        const int n  = lane & 15;
        const int kh = lane >> 4;
        const int d0 = (2 * wave) * 16;     // tile A for this wave
        const int d1 = d0 + 16;             // tile B for this wave
        v8f acc0 = {};
        v8f acc1 = {};
        #pragma unroll
        for (int ch = 0; ch < 13; ++ch) {
            int kk = 4 * ch + 2 * kh;
            v2f a, b0, b1;
            a[0]  = sW[kk];
            a[1]  = sW[kk + 1];
            b0[0] = sV[kk * ROW + d0 + n];
            b0[1] = sV[(kk + 1) * ROW + d0 + n];
            b1[0] = sV[kk * ROW + d1 + n];
            b1[1] = sV[(kk + 1) * ROW + d1 + n];
            acc0 = __builtin_amdgcn_wmma_f32_16x16x4_f32(false, a, false, b0,
                                                         (short)0, acc0, false, false);
            acc1 = __builtin_amdgcn_wmma_f32_16x16x4_f32(false, a, false, b1,
                                                         (short)0, acc1, false, false);
        }
        // All 16 M-rows identical: lanes 0-15 hold tile-A result (M=0) in acc0[0],
        // lanes 16-31 hold tile-B result (M=8) in acc1[0] -> full-wave store.
        float u = (kh == 0) ? acc0[0] : acc1[0];
        int   d = (kh == 0) ? (d0 + n) : (d1 + n);
        out[(size_t)b * D_DIM + d] = u;
    }
}

extern "C" void kernel_launch(void* const* d_in, const int* in_sizes, int n_in,
                              void* d_out, int out_size, void* d_ws, size_t ws_size,
                              hipStream_t stream) {
    const float*     interest = (const float*)d_in[0];     // [bs, 50, 256] f32
    const float*     cand     = (const float*)d_in[1];     // [bs, 256] f32
    const long long* counts   = (const long long*)d_in[2]; // [bs] i64
    float*           out      = (float*)d_out;             // [bs, 256] f32
    const int bs = in_sizes[2];
    ica_kernel<<<dim3(bs), dim3(256), 0, stream>>>(interest, cand, counts, out);
}